// NextIntegerLinkPredictor_15522011808327
// MI455X (gfx1250) — compile-verified
//
#include <hip/hip_runtime.h>
#include <hip/hip_bf16.h>
#include <math.h>

// ---------------------------------------------------------------------------
// CDNA5 (gfx1250) f32 WMMA: D = A(16x4) * B(4x16) + C(16x16), wave32.
// Per-lane fragments (ISA 7.12.2):
//   A: lane l<16 -> row M=l,  v0=K0,v1=K1 ; lane l>=16 -> row M=l-16, v0=K2,v1=K3
//   B: lane l<16 -> col N=l,  v0=K0,v1=K1 ; lane l>=16 -> col N=l-16, v0=K2,v1=K3
//   C/D: VGPR r : lanes 0-15 -> (M=r, N=lane), lanes 16-31 -> (M=r+8, N=lane-16)
// ---------------------------------------------------------------------------
typedef float v2f __attribute__((ext_vector_type(2)));
typedef float v8f __attribute__((ext_vector_type(8)));

__device__ __forceinline__ v8f wmma16x16x4(v2f a, v2f b, v8f c) {
#if !defined(__HIP_DEVICE_COMPILE__)
  (void)a; (void)b;
  return c; // host stub, never executed
#elif __has_builtin(__builtin_amdgcn_wmma_f32_16x16x4_f32)
  // 8 args: (neg_a, A, neg_b, B, c_mod, C, reuse_a, reuse_b)
  return __builtin_amdgcn_wmma_f32_16x16x4_f32(false, a, false, b, (short)0, c,
                                               false, false);
#else
#error "__builtin_amdgcn_wmma_f32_16x16x4_f32 not available for this target"
#endif
}

// ---------------------------------------------------------------------------
// Degree / normalization kernels
// ---------------------------------------------------------------------------
__global__ void init_deg_kernel(float* __restrict__ deg, int n) {
  int i = blockIdx.x * blockDim.x + threadIdx.x;
  if (i < n) deg[i] = 1.0f;  // self-loop counts as one incoming edge
}

__global__ void deg_edges_kernel(const int* __restrict__ dst,
                                 float* __restrict__ deg, int E) {
  int e = blockIdx.x * blockDim.x + threadIdx.x;
  if (e < E) atomicAdd(&deg[dst[e]], 1.0f);
}

__global__ void finalize_dinv_kernel(float* __restrict__ deg, int n) {
  int i = blockIdx.x * blockDim.x + threadIdx.x;
  if (i < n) {
    float d = deg[i];
    deg[i] = (d > 0.0f) ? rsqrtf(d) : 0.0f;  // in-place deg -> dinv
  }
}

// ---------------------------------------------------------------------------
// GEMM via f32 WMMA with LDS staging:
//   C[M x Nout] = A[M x K] @ B[K x Nout] (+bias)(+relu)
// Block: 256 threads = 8 waves. Block tile: 128 rows x 16 cols.
// K is consumed in chunks of BK=64:
//   - A chunk (128 x 64) loaded coalesced as b128 into As (row stride 68 pads
//     fragment reads onto 32 distinct LDS banks).
//   - B chunk loaded transposed into Bs[n][kk] so the per-lane fragment read
//     is a contiguous, conflict-free ds_load_b64.
// Requires K % 64 == 0, Nout % 16 == 0 (true here: K in {128,256}).
// ---------------------------------------------------------------------------
#define BK 64
#define APAD (BK + 4) /* 68-float row stride: bank = (4*r + c) % 64 */

__global__ __launch_bounds__(256) void gemm_wmma_kernel(
    const float* __restrict__ A, const float* __restrict__ B,
    const float* __restrict__ bias, float* __restrict__ C, int M, int K,
    int Nout, int do_relu) {
  __shared__ float As[128 * APAD];
  __shared__ float Bs[16 * APAD];

  const int tid = threadIdx.x;
  const int lane = tid & 31;
  const int wv = tid >> 5;
  const int half = lane >> 4;  // 0: lanes 0-15, 1: lanes 16-31
  const int lid = lane & 15;

  const int rowbase = blockIdx.y * 128;
  const int row0 = rowbase + wv * 16;
  const int col0 = blockIdx.x * 16;
  const int col = col0 + lid;
  const bool active = (row0 < M);  // wave-uniform

  v8f acc = {0.f, 0.f, 0.f, 0.f, 0.f, 0.f, 0.f, 0.f};

  for (int k0 = 0; k0 < K; k0 += BK) {
    // --- A chunk: 128 rows x 64 cols, b128 coalesced (2048 float4 / block).
    for (int i = tid; i < 128 * (BK / 4); i += 256) {
      const int r = i >> 4;            // 0..127
      const int c4 = (i & 15) << 2;    // 0,4,...,60
      int rg = rowbase + r;
      if (rg >= M) rg = M - 1;  // clamp (safe duplicate; rows >= M unused)
      const float* gp = &A[(size_t)rg * K + k0 + c4];
      *(float4*)&As[r * APAD + c4] = *(const float4*)gp;
      if (k0 + BK < K) __builtin_prefetch(gp + BK, 0, 0);  // global_prefetch
    }
    // --- B chunk, transposed: Bs[n][kk] = B[k0+kk][col0+n].
    for (int i = tid; i < 16 * BK; i += 256) {
      const int kk = i >> 4;
      const int n = i & 15;
      Bs[n * APAD + kk] = B[(size_t)(k0 + kk) * Nout + col0 + n];
    }
    __syncthreads();

    if (active) {
      const float* __restrict__ ap = &As[(wv * 16 + lid) * APAD + 2 * half];
      const float* __restrict__ bp = &Bs[lid * APAD + 2 * half];
#pragma unroll
      for (int kk = 0; kk < BK; kk += 4) {
        v2f a = *(const v2f*)(ap + kk);  // ds_load_b64, conflict-free
        v2f b = *(const v2f*)(bp + kk);  // ds_load_b64, conflict-free
        acc = wmma16x16x4(a, b, acc);
      }
    }
    __syncthreads();
  }

  if (active) {
    const float bv = bias ? bias[col] : 0.0f;
#pragma unroll
    for (int r = 0; r < 8; ++r) {
      const int row = row0 + r + 8 * half;
      float v = acc[r] + bv;
      if (do_relu) v = fmaxf(v, 0.0f);
      C[(size_t)row * Nout + col] = v;
    }
  }
}

// ---------------------------------------------------------------------------
// GCN aggregation: agg = self-loop term, then atomic scatter over edges.
// ---------------------------------------------------------------------------
__global__ void self_init_kernel(const float* __restrict__ hw,
                                 const float* __restrict__ dinv,
                                 float* __restrict__ agg, int n, int F) {
  int g = blockIdx.x * blockDim.x + threadIdx.x;
  if (g < n * F) {
    float w = dinv[g / F];
    agg[g] = hw[g] * w * w;  // self-loop: norm = dinv_i * dinv_i
  }
}

__global__ __launch_bounds__(256) void edge_scatter_kernel(
    const float* __restrict__ hw, const int* __restrict__ src,
    const int* __restrict__ dst, const float* __restrict__ dinv,
    float* __restrict__ agg, int E, int F) {
  const int e = blockIdx.x * (blockDim.x >> 5) + (threadIdx.x >> 5);
  if (e >= E) return;
  const int lane = threadIdx.x & 31;
  const int s = src[e];
  const int d = dst[e];
  const float w = dinv[s] * dinv[d];
  const float* __restrict__ hs = hw + (size_t)s * F;
  float* __restrict__ ad = agg + (size_t)d * F;
  for (int f = lane; f < F; f += 32) atomicAdd(&ad[f], hs[f] * w);
}

__global__ void bias_relu_kernel(const float* __restrict__ agg,
                                 const float* __restrict__ b,
                                 float* __restrict__ h, int total, int F) {
  int g = blockIdx.x * blockDim.x + threadIdx.x;
  if (g < total) h[g] = fmaxf(agg[g] + b[g % F], 0.0f);
}

// ---------------------------------------------------------------------------
// Fused decoder: 128 threads (4 waves) per 16 query edges; waves split the
// output N-tiles of each layer.
//   zc = [z[qs] | z[qd] | int_diff | same_part]           (16 x 258)
//   u1 = relu(zc @ Wd1 + bd1)                             (16 x 256)
//   u2 = relu(u1 @ Wd2 + bd2)                             (16 x 128)
//   out = sigmoid(u2 @ Wd3 + bd3)                         (16)
// The 2 scalar features are a rank-2 VALU update (WMMA K stays 256).
// LDS row strides padded (260 / 132) so fragment ds_load_b64s are
// bank-conflict free: bank = (4*row + col) % 64.
// Hardcoded dims: D=128, 2D=256, H1=256, H2=128 (per reference).
// ---------------------------------------------------------------------------
#define ZS 260  /* zc / u1 row stride (floats) */
#define US 132  /* u2 row stride (floats) */

__global__ __launch_bounds__(128) void decoder_kernel(
    const float* __restrict__ z, const int* __restrict__ qs,
    const int* __restrict__ qd, const int* __restrict__ iv,
    const int* __restrict__ pid, const float* __restrict__ Wd1,
    const float* __restrict__ bd1, const float* __restrict__ Wd2,
    const float* __restrict__ bd2, const float* __restrict__ Wd3,
    const float* __restrict__ bd3, float* __restrict__ out, int Q) {
  __shared__ float zc[16 * ZS];  // layer-1 A operand; reused as u2 (16 x US)
  __shared__ float u1[16 * ZS];  // layer-1 output / layer-2 A operand
  __shared__ int sidx[16], didx[16];
  __shared__ float sdif[16], spar[16];

  const int tid = threadIdx.x;
  const int lane = tid & 31;
  const int wv = tid >> 5;
  const int half = lane >> 4;
  const int lid = lane & 15;
  const int qbase = blockIdx.x * 16;

  if (tid < 16) {
    int q = qbase + tid;
    if (q >= Q) q = Q - 1;  // clamp (dims are exact multiples; safety only)
    const int s = qs[q];
    const int d = qd[q];
    sidx[tid] = s;
    didx[tid] = d;
    sdif[tid] = (float)(iv[d] - iv[s]);
    spar[tid] = (pid[s] == pid[d]) ? 1.0f : 0.0f;
  }
  __syncthreads();

  // Gather z[qs], z[qd] rows (128 f32 each) into zc as b128 loads.
  for (int i = tid; i < 16 * 32; i += 128) {
    const int r = i >> 5;
    const int f4 = (i & 31) << 2;  // 0,4,...,124
    *(float4*)&zc[r * ZS + f4] =
        *(const float4*)&z[(size_t)sidx[r] * 128 + f4];
    *(float4*)&zc[r * ZS + 128 + f4] =
        *(const float4*)&z[(size_t)didx[r] * 128 + f4];
  }
  __syncthreads();

  // Layer 1: u1 = relu(zc(16x256) @ Wd1[0:256,:] + rank2 + bd1).
  for (int nt = wv; nt < 16; nt += 4) {
    const int col = nt * 16 + lid;
    v8f acc = {0.f, 0.f, 0.f, 0.f, 0.f, 0.f, 0.f, 0.f};
    const float* __restrict__ ap = &zc[lid * ZS + 2 * half];
    for (int k = 0; k < 256; k += 4) {
      v2f a = *(const v2f*)(ap + k);  // ds_load_b64, conflict-free
      v2f b;
      b.x = Wd1[(size_t)(k + 2 * half) * 256 + col];
      b.y = Wd1[(size_t)(k + 2 * half + 1) * 256 + col];
      acc = wmma16x16x4(a, b, acc);
    }
    const float w256 = Wd1[(size_t)256 * 256 + col];  // int_diff row
    const float w257 = Wd1[(size_t)257 * 256 + col];  // same_part row
    const float bb = bd1[col];
#pragma unroll
    for (int r = 0; r < 8; ++r) {
      const int row = r + 8 * half;
      float v = acc[r] + sdif[row] * w256 + spar[row] * w257 + bb;
      u1[row * ZS + col] = fmaxf(v, 0.0f);
    }
  }
  __syncthreads();

  // Layer 2: u2 = relu(u1(16x256) @ Wd2(256x128) + bd2); u2 overlays zc.
  float* __restrict__ u2 = zc;
  for (int nt = wv; nt < 8; nt += 4) {
    const int col = nt * 16 + lid;
    v8f acc = {0.f, 0.f, 0.f, 0.f, 0.f, 0.f, 0.f, 0.f};
    const float* __restrict__ ap = &u1[lid * ZS + 2 * half];
    for (int k = 0; k < 256; k += 4) {
      v2f a = *(const v2f*)(ap + k);
      v2f b;
      b.x = Wd2[(size_t)(k + 2 * half) * 128 + col];
      b.y = Wd2[(size_t)(k + 2 * half + 1) * 128 + col];
      acc = wmma16x16x4(a, b, acc);
    }
    const float bb = bd2[col];
#pragma unroll
    for (int r = 0; r < 8; ++r) {
      const int row = r + 8 * half;
      u2[row * US + col] = fmaxf(acc[r] + bb, 0.0f);
    }
  }
  __syncthreads();

  // Layer 3: 128 -> 1 dot product + sigmoid, one lane per query row.
  if (tid < 16) {
    const int q = qbase + tid;
    if (q < Q) {
      float s = bd3[0];
      for (int f = 0; f < 128; ++f) s += u2[tid * US + f] * Wd3[f];
      out[q] = 1.0f / (1.0f + __expf(-s));
    }
  }
}

// ---------------------------------------------------------------------------
// Launcher
// ---------------------------------------------------------------------------
extern "C" void kernel_launch(void* const* d_in, const int* in_sizes, int n_in,
                              void* d_out, int out_size, void* d_ws,
                              size_t ws_size, hipStream_t stream) {
  const float* x    = (const float*)d_in[0];
  const int*  edge  = (const int*)d_in[1];
  const int*  qedge = (const int*)d_in[2];
  const int*  iv    = (const int*)d_in[3];
  const int*  pid   = (const int*)d_in[4];
  const float* W1  = (const float*)d_in[5];
  const float* b1  = (const float*)d_in[6];
  const float* W2  = (const float*)d_in[7];
  const float* b2  = (const float*)d_in[8];
  const float* Wfc = (const float*)d_in[9];
  const float* bfc = (const float*)d_in[10];
  const float* Wd1 = (const float*)d_in[11];
  const float* bd1 = (const float*)d_in[12];
  const float* Wd2 = (const float*)d_in[13];
  const float* bd2 = (const float*)d_in[14];
  const float* Wd3 = (const float*)d_in[15];
  const float* bd3 = (const float*)d_in[16];

  const int N = in_sizes[3];          // nodes
  const int E = in_sizes[1] / 2;      // edges
  const int Q = in_sizes[2] / 2;      // query edges
  const int F = in_sizes[0] / N;      // 128
  const int H = in_sizes[6];          // 256
  const int D = in_sizes[10];         // 128

  const int* esrc = edge;
  const int* edst = edge + E;
  const int* qs = qedge;
  const int* qd = qedge + Q;

  // Workspace layout (floats): dinv[N] | hw[N*H] | agg[N*H] | hbuf[N*H] | z[N*D]
  float* ws   = (float*)d_ws;
  float* dinv = ws;
  float* hw   = dinv + N;
  float* agg  = hw + (size_t)N * H;
  float* hbuf = agg + (size_t)N * H;
  float* zbuf = hbuf + (size_t)N * H;
  float* outp = (float*)d_out;

  const dim3 b256(256);
  const int nh = N * H;

  // Degree normalization (self-loop included via init to 1.0).
  init_deg_kernel<<<(N + 255) / 256, b256, 0, stream>>>(dinv, N);
  deg_edges_kernel<<<(E + 255) / 256, b256, 0, stream>>>(edst, dinv, E);
  finalize_dinv_kernel<<<(N + 255) / 256, b256, 0, stream>>>(dinv, N);

  // GCN layer 1: hw = x @ W1 ; aggregate ; h1 = relu(agg + b1)
  gemm_wmma_kernel<<<dim3(H / 16, (N + 127) / 128), b256, 0, stream>>>(
      x, W1, nullptr, hw, N, F, H, 0);
  self_init_kernel<<<(nh + 255) / 256, b256, 0, stream>>>(hw, dinv, agg, N, H);
  edge_scatter_kernel<<<(E + 7) / 8, b256, 0, stream>>>(hw, esrc, edst, dinv,
                                                        agg, E, H);
  bias_relu_kernel<<<(nh + 255) / 256, b256, 0, stream>>>(agg, b1, hbuf, nh, H);

  // GCN layer 2: hw = h1 @ W2 ; aggregate ; h2 = relu(agg + b2)
  gemm_wmma_kernel<<<dim3(H / 16, (N + 127) / 128), b256, 0, stream>>>(
      hbuf, W2, nullptr, hw, N, H, H, 0);
  self_init_kernel<<<(nh + 255) / 256, b256, 0, stream>>>(hw, dinv, agg, N, H);
  edge_scatter_kernel<<<(E + 7) / 8, b256, 0, stream>>>(hw, esrc, edst, dinv,
                                                        agg, E, H);
  bias_relu_kernel<<<(nh + 255) / 256, b256, 0, stream>>>(agg, b2, hbuf, nh, H);

  // Encoder FC: z = h2 @ Wfc + bfc
  gemm_wmma_kernel<<<dim3(D / 16, (N + 127) / 128), b256, 0, stream>>>(
      hbuf, Wfc, bfc, zbuf, N, H, D, 0);

  // Fused decoder MLP over query edges.
  decoder_kernel<<<(Q + 15) / 16, dim3(128), 0, stream>>>(
      zbuf, qs, qd, iv, pid, Wd1, bd1, Wd2, bd2, Wd3, bd3, outp, Q);
}